// Coref_34995393527957
// MI455X (gfx1250) — compile-verified
//
#include <hip/hip_runtime.h>
#include <hip/hip_bf16.h>
#include <math.h>

typedef __attribute__((ext_vector_type(2))) float v2f;
typedef __attribute__((ext_vector_type(8))) float v8f;

#define T_LEN 768
#define D_EMB 300
#define H_LSTM 200
#define SD_ 400
#define HID_ 150
#define G4_ 800
#define NS_ 3066
#define KK_ 307
#define SPAN_ 1101
#define PAIRW_ 3303
// padded dims (all WMMA operands guard-free)
#define LDK_ 1104   // SPAN_ padded to mult of 4 (and 16)
#define NH_ 160     // HID_ padded to mult of 16
#define NSP_ 3072   // NS_ padded to mult of 16
#define KKP_ 320    // KK_ padded to mult of 16

// ---------------- workspace layout (floats) ----------------
constexpr size_t OF_EMB  = 0;
constexpr size_t OF_PREF = OF_EMB  + (size_t)T_LEN * D_EMB;
constexpr size_t OF_PRER = OF_PREF + (size_t)T_LEN * G4_;
constexpr size_t OF_ST   = OF_PRER + (size_t)T_LEN * G4_;
constexpr size_t OF_T1   = OF_ST   + (size_t)T_LEN * SD_;
constexpr size_t OF_T2   = OF_T1   + (size_t)NSP_ * NH_;
constexpr size_t OF_A    = OF_T2   + (size_t)NSP_ * NH_;
constexpr size_t OF_G    = OF_A    + (size_t)T_LEN;
constexpr size_t OF_SM   = OF_G    + (size_t)NSP_ * LDK_;
constexpr size_t OF_GK   = OF_SM   + (size_t)NS_;
constexpr size_t OF_SMK  = OF_GK   + (size_t)KKP_ * LDK_;
constexpr size_t OF_AM   = OF_SMK  + (size_t)KK_;
constexpr size_t OF_BM   = OF_AM   + (size_t)KKP_ * NH_;
constexpr size_t OF_LG   = OF_BM   + (size_t)KKP_ * NH_;
// padded weight copies
constexpr size_t OF_AW1  = OF_LG   + (size_t)KK_ * KK_;
constexpr size_t OF_AW2  = OF_AW1  + (size_t)NH_ * SD_;
constexpr size_t OF_SW1  = OF_AW2  + (size_t)NH_ * NH_;
constexpr size_t OF_SW2  = OF_SW1  + (size_t)NH_ * LDK_;
constexpr size_t OF_WA   = OF_SW2  + (size_t)NH_ * NH_;
constexpr size_t OF_WB   = OF_WA   + (size_t)NH_ * LDK_;
constexpr size_t OF_WC   = OF_WB   + (size_t)NH_ * LDK_;
constexpr size_t OF_PW2  = OF_WC   + (size_t)NH_ * LDK_;
constexpr size_t OF_AB1  = OF_PW2  + (size_t)NH_ * NH_;
constexpr size_t OF_AB2  = OF_AB1  + NH_;
constexpr size_t OF_SB1  = OF_AB2  + NH_;
constexpr size_t OF_SB2  = OF_SB1  + NH_;
constexpr size_t OF_PB1  = OF_SB2  + NH_;
constexpr size_t OF_PB2  = OF_PB1  + NH_;
constexpr size_t OF_PW3  = OF_PB2  + NH_;
constexpr size_t OF_INT  = OF_PW3  + NH_;

__device__ __forceinline__ int imin(int a, int b) { return a < b ? a : b; }

__device__ __forceinline__ void span_decode(int p, int& s, int& w) {
    if (p < 768)       { w = 1; s = p; }
    else if (p < 1535) { w = 2; s = p - 768; }
    else if (p < 2301) { w = 3; s = p - 1535; }
    else               { w = 4; s = p - 2301; }
}

__device__ __forceinline__ float sigm(float x) { return 1.f / (1.f + expf(-x)); }

// ---------------- zero-padded copy: dst[r*ldd+c] = src[r*lds+c] (in-range) else 0 ----------------
__global__ void padcopy_kernel(const float* __restrict__ src, int lds, int rows, int cols,
                               float* __restrict__ dst, int ldd, int rowsP) {
    int idx = blockIdx.x * blockDim.x + threadIdx.x;
    if (idx >= rowsP * ldd) return;
    int r = idx / ldd, c = idx - r * ldd;
    dst[idx] = (r < rows && c < cols) ? src[(size_t)r * lds + c] : 0.f;
}

// ---------------- embedding gather ----------------
__global__ void gather_embed_kernel(const int* __restrict__ idx,
                                    const float* __restrict__ embed,
                                    float* __restrict__ emb) {
    int t = blockIdx.x;
    size_t row = (size_t)idx[t] * D_EMB;
    for (int d = threadIdx.x; d < D_EMB; d += blockDim.x)
        emb[(size_t)t * D_EMB + d] = embed[row + d];
}

// ---------------- guard-free fp32 WMMA GEMM: C = act(A @ B^T + b1 + b2) ----------------
// Requirements: M mult 16, N mult 16, K mult 4, operands zero-padded.
__global__ void __launch_bounds__(256) gemm_wmma_kernel(
    const float* __restrict__ A, int lda,
    const float* __restrict__ B, int ldb,
    float* __restrict__ C, int ldc,
    int N, int K,
    const float* __restrict__ bias1,
    const float* __restrict__ bias2,
    int relu)
{
    int wave = threadIdx.x >> 5;
    int lane = threadIdx.x & 31;
    int half = lane >> 4;
    int r    = lane & 15;
    int m0 = blockIdx.x * 16;
    int n0 = blockIdx.y * 128 + wave * 16;
    if (n0 >= N) return;   // uniform per wave

    const float* Ap = A + (size_t)(m0 + r) * lda + 2 * half;
    const float* Bp = B + (size_t)(n0 + r) * ldb + 2 * half;

    v8f acc = {0.f,0.f,0.f,0.f,0.f,0.f,0.f,0.f};
    for (int k0 = 0; k0 < K; k0 += 4) {
        v2f a = *(const v2f*)(Ap + k0);
        v2f b = *(const v2f*)(Bp + k0);
        acc = __builtin_amdgcn_wmma_f32_16x16x4_f32(false, a, false, b,
                                                    (short)0, acc, false, false);
    }

    int n = n0 + r;
    float bv = (bias1 ? bias1[n] : 0.f) + (bias2 ? bias2[n] : 0.f);
#pragma unroll
    for (int e = 0; e < 8; ++e) {
        int m = m0 + e + 8 * half;
        float v = acc[e] + bv;
        if (relu) v = fmaxf(v, 0.f);
        C[(size_t)m * ldc + n] = v;
    }
}

// ---------------- out[m] = A[m,:] . w + b ----------------
__global__ void rowdot_kernel(const float* __restrict__ A, int lda,
                              const float* __restrict__ w,
                              const float* __restrict__ b,
                              float* __restrict__ out, int M, int K) {
    int m = blockIdx.x * blockDim.x + threadIdx.x;
    if (m >= M) return;
    const float* ar = A + (size_t)m * lda;
    float s = b[0];
    for (int k = 0; k < K; ++k) s += ar[k] * w[k];
    out[m] = s;
}

// ---------------- sequential bi-LSTM (block 0 = fwd, block 1 = rev) ----------------
__global__ void __launch_bounds__(256) lstm_kernel(
    const float* __restrict__ preF, const float* __restrict__ preR,
    const float* __restrict__ whhF, const float* __restrict__ whhR,
    float* __restrict__ states)
{
    int dir = blockIdx.x;
    const float* pre = dir ? preR : preF;
    const float* whh = dir ? whhR : whhF;
    __shared__ float h[H_LSTM], c[H_LSTM], gates[G4_];
    int tid = threadIdx.x;
    if (tid < H_LSTM) { h[tid] = 0.f; c[tid] = 0.f; }
    __syncthreads();
    for (int step = 0; step < T_LEN; ++step) {
        int t = dir ? (T_LEN - 1 - step) : step;
        for (int j = tid; j < G4_; j += 256) {
            float s = pre[(size_t)t * G4_ + j];
            const float* wr = whh + (size_t)j * H_LSTM;
            for (int k = 0; k < H_LSTM; ++k) s += wr[k] * h[k];
            gates[j] = s;
        }
        __syncthreads();
        if (tid < H_LSTM) {
            float ig = sigm(gates[tid]);
            float fg = sigm(gates[H_LSTM + tid]);
            float gg = tanhf(gates[2 * H_LSTM + tid]);
            float og = sigm(gates[3 * H_LSTM + tid]);
            float cn = fg * c[tid] + ig * gg;
            c[tid] = cn;
            float hn = og * tanhf(cn);
            h[tid] = hn;
            states[(size_t)t * SD_ + dir * H_LSTM + tid] = hn;
        }
        __syncthreads();
    }
}

// ---------------- span features g[p, 0:1104] (rows/cols padded with zeros) ----------------
__global__ void span_kernel(const float* __restrict__ a,
                            const float* __restrict__ emb,
                            const float* __restrict__ states,
                            float* __restrict__ g) {
    int p = blockIdx.x;
    float* gr = g + (size_t)p * LDK_;
    if (p >= NS_) {   // zero pad rows
        for (int d = threadIdx.x; d < LDK_; d += blockDim.x) gr[d] = 0.f;
        return;
    }
    int s, w; span_decode(p, s, w);
    int e = s + w - 1;
    __shared__ float attw[4];
    if (threadIdx.x == 0) {
        float lo[4], mx = -1e30f;
        for (int o = 0; o < 4; ++o) {
            lo[o] = (o < w) ? a[imin(s + o, T_LEN - 1)] : -1e30f;
            mx = fmaxf(mx, lo[o]);
        }
        float ex[4], sum = 0.f;
        for (int o = 0; o < 4; ++o) { ex[o] = expf(lo[o] - mx); sum += ex[o]; }
        for (int o = 0; o < 4; ++o) attw[o] = ex[o] / sum;
    }
    __syncthreads();
    for (int d = threadIdx.x; d < SD_; d += blockDim.x) {
        gr[d]       = states[(size_t)s * SD_ + d];
        gr[SD_ + d] = states[(size_t)e * SD_ + d];
    }
    for (int d = threadIdx.x; d < D_EMB; d += blockDim.x) {
        float acc = 0.f;
        for (int o = 0; o < 4; ++o)
            acc += attw[o] * emb[(size_t)imin(s + o, T_LEN - 1) * D_EMB + d];
        gr[2 * SD_ + d] = acc;
    }
    if (threadIdx.x == 0) {
        gr[SPAN_ - 1] = (float)w;
        gr[SPAN_]     = 0.f;   // pad cols 1101..1103
        gr[SPAN_ + 1] = 0.f;
        gr[SPAN_ + 2] = 0.f;
    }
}

// ---------------- top-k by rank counting ----------------
__global__ void rank_kernel(const float* __restrict__ sm, int* __restrict__ keep) {
    int i = blockIdx.x * blockDim.x + threadIdx.x;
    if (i >= NS_) return;
    float si = sm[i];
    int rank = 0;
    for (int j = 0; j < NS_; ++j) {
        float sj = sm[j];
        rank += (sj > si) || (sj == si && j < i);
    }
    keep[i] = (rank < KK_) ? 1 : 0;
}

__global__ void scatter_kernel(const int* __restrict__ keep, int* __restrict__ kidx) {
    int i = blockIdx.x * blockDim.x + threadIdx.x;
    if (i >= NS_ || !keep[i]) return;
    int s, w; span_decode(i, s, w);
    int key_i = s * T_LEN + (s + w - 1);
    int pos = 0;
    for (int j = 0; j < NS_; ++j) {
        if (!keep[j]) continue;
        int s2, w2; span_decode(j, s2, w2);
        pos += ((s2 * T_LEN + (s2 + w2 - 1)) < key_i);
    }
    kidx[pos] = i;
}

__global__ void gatherG_kernel(const float* __restrict__ g,
                               const float* __restrict__ sm,
                               const int* __restrict__ kidx,
                               float* __restrict__ Gk, float* __restrict__ smk,
                               float* __restrict__ out) {
    int rI = blockIdx.x;
    float* dst = Gk + (size_t)rI * LDK_;
    if (rI >= KK_) {   // zero pad rows 307..319
        for (int d = threadIdx.x; d < LDK_; d += blockDim.x) dst[d] = 0.f;
        return;
    }
    int p = kidx[rI];
    const float* src = g + (size_t)p * LDK_;
    for (int d = threadIdx.x; d < LDK_; d += blockDim.x) dst[d] = src[d];
    if (threadIdx.x == 0) {
        smk[rI] = sm[p];
        int s, w; span_decode(p, s, w);
        out[(size_t)KK_ * (KK_ + 1) + rI]       = (float)s;
        out[(size_t)KK_ * (KK_ + 1) + KK_ + rI] = (float)(s + w - 1);
    }
}

// ---------------- fused pair scorer (all operands padded; guard-free WMMA loops) ----------------
__global__ void __launch_bounds__(128) pair_kernel(
    const float* __restrict__ G,    // KKP_ x LDK_
    const float* __restrict__ Amat, // KKP_ x NH_
    const float* __restrict__ Bmat, // KKP_ x NH_
    const float* __restrict__ smk,
    const float* __restrict__ wcp,  // NH_ x LDK_
    const float* __restrict__ pb1p, // NH_
    const float* __restrict__ pw2p, // NH_ x NH_
    const float* __restrict__ pb2p, // NH_
    const float* __restrict__ pw3p, // NH_
    const float* __restrict__ pb3,
    float* __restrict__ lg)
{
    __shared__ float h1s[4][16 * NH_];
    int wave = threadIdx.x >> 5, lane = threadIdx.x & 31;
    int half = lane >> 4, r = lane & 15;
    int i0 = blockIdx.x * 16;
    int j  = blockIdx.y * 4 + wave;
    if (j >= KK_) return;   // wave-uniform

    int irow = i0 + r;

    const float* Gi = G + (size_t)irow * LDK_ + 2 * half;
    const float* Gj = G + (size_t)j * LDK_ + 2 * half;
    const float* Wc = wcp + (size_t)r * LDK_ + 2 * half;

    v8f acc[10];
#pragma unroll
    for (int nt = 0; nt < 10; ++nt) acc[nt] = (v8f){0.f,0.f,0.f,0.f,0.f,0.f,0.f,0.f};

    // layer-1: X[p,k] = G[i,k]*G[j,k], X @ wc^T  (K = 1104 padded)
    for (int k0 = 0; k0 < LDK_; k0 += 4) {
        v2f ga = *(const v2f*)(Gi + k0);
        v2f gb = *(const v2f*)(Gj + k0);
        v2f a = ga * gb;
#pragma unroll
        for (int nt = 0; nt < 10; ++nt) {
            v2f b = *(const v2f*)(Wc + (size_t)nt * 16 * LDK_ + k0);
            acc[nt] = __builtin_amdgcn_wmma_f32_16x16x4_f32(false, a, false, b,
                                                            (short)0, acc[nt], false, false);
        }
    }

    // h1 = relu(C + A_i + B_j + pb1) -> LDS (16 x 160)
    float* hw = h1s[wave];
#pragma unroll
    for (int nt = 0; nt < 10; ++nt) {
        int n = nt * 16 + r;
        float bj = pb1p[n] + Bmat[(size_t)j * NH_ + n];
#pragma unroll
        for (int e = 0; e < 8; ++e) {
            int m = e + 8 * half;
            float av = Amat[(size_t)(i0 + m) * NH_ + n];
            hw[m * NH_ + n] = fmaxf(acc[nt][e] + av + bj, 0.f);
        }
    }

    // layer-2: h1 @ pw2p^T  (K = 160 padded, pads are zero on both sides)
    const float* Hm = hw + r * NH_ + 2 * half;
    const float* W2 = pw2p + (size_t)r * NH_ + 2 * half;
    v8f acc2[10];
#pragma unroll
    for (int nt = 0; nt < 10; ++nt) acc2[nt] = (v8f){0.f,0.f,0.f,0.f,0.f,0.f,0.f,0.f};
    for (int k0 = 0; k0 < NH_; k0 += 4) {
        v2f a = *(const v2f*)(Hm + k0);
#pragma unroll
        for (int nt = 0; nt < 10; ++nt) {
            v2f b = *(const v2f*)(W2 + (size_t)nt * 16 * NH_ + k0);
            acc2[nt] = __builtin_amdgcn_wmma_f32_16x16x4_f32(false, a, false, b,
                                                             (short)0, acc2[nt], false, false);
        }
    }

    // h2 = relu(acc2 + pb2) -> LDS (reuse), then sa = h2 . pw3
#pragma unroll
    for (int nt = 0; nt < 10; ++nt) {
        int n = nt * 16 + r;
        float bv = pb2p[n];
#pragma unroll
        for (int e = 0; e < 8; ++e) {
            int m = e + 8 * half;
            hw[m * NH_ + n] = fmaxf(acc2[nt][e] + bv, 0.f);
        }
    }
    float sa = 0.f;
    int nb = half * 80;
    for (int n = nb; n < nb + 80; ++n) sa += hw[r * NH_ + n] * pw3p[n];
    sa += __shfl_xor(sa, 16, 32);
    if (half == 0 && irow < KK_) {
        float full = smk[irow] + smk[j] + sa + pb3[0];
        lg[(size_t)irow * KK_ + j] = (irow > j) ? full : -1e30f;
    }
}

// ---------------- row softmax over [lg row | 0] -> pred (row stride KK_+1) ----------------
__global__ void softmax_kernel(const float* __restrict__ lg, float* __restrict__ pred) {
    int row = blockIdx.x;
    __shared__ float red[128];
    int tid = threadIdx.x;
    float mx = 0.f;   // appended zero column participates in the max
    for (int j = tid; j < KK_; j += 128) mx = fmaxf(mx, lg[(size_t)row * KK_ + j]);
    red[tid] = mx; __syncthreads();
    for (int s = 64; s > 0; s >>= 1) { if (tid < s) red[tid] = fmaxf(red[tid], red[tid + s]); __syncthreads(); }
    mx = red[0]; __syncthreads();
    float sum = (tid == 0) ? expf(0.f - mx) : 0.f;
    for (int j = tid; j < KK_; j += 128) sum += expf(lg[(size_t)row * KK_ + j] - mx);
    red[tid] = sum; __syncthreads();
    for (int s = 64; s > 0; s >>= 1) { if (tid < s) red[tid] += red[tid + s]; __syncthreads(); }
    float inv = 1.f / red[0];
    for (int j = tid; j < KK_; j += 128)
        pred[(size_t)row * (KK_ + 1) + j] = expf(lg[(size_t)row * KK_ + j] - mx) * inv;
    if (tid == 0) pred[(size_t)row * (KK_ + 1) + KK_] = expf(0.f - mx) * inv;
}

// ---------------- host launcher ----------------
static inline void padcopy(const float* src, int lds, int rows, int cols,
                           float* dst, int ldd, int rowsP, hipStream_t stream) {
    int total = rowsP * ldd;
    padcopy_kernel<<<(total + 255) / 256, 256, 0, stream>>>(src, lds, rows, cols, dst, ldd, rowsP);
}

extern "C" void kernel_launch(void* const* d_in, const int* in_sizes, int n_in,
                              void* d_out, int out_size, void* d_ws, size_t ws_size,
                              hipStream_t stream) {
    (void)in_sizes; (void)n_in; (void)out_size; (void)ws_size;
    const int*   tok    = (const int*)  d_in[0];
    const float* embed  = (const float*)d_in[1];
    const float* w_ih_f = (const float*)d_in[2];
    const float* w_hh_f = (const float*)d_in[3];
    const float* b_ih_f = (const float*)d_in[4];
    const float* b_hh_f = (const float*)d_in[5];
    const float* w_ih_r = (const float*)d_in[6];
    const float* w_hh_r = (const float*)d_in[7];
    const float* b_ih_r = (const float*)d_in[8];
    const float* b_hh_r = (const float*)d_in[9];
    const float* aw1 = (const float*)d_in[10]; const float* ab1 = (const float*)d_in[11];
    const float* aw2 = (const float*)d_in[12]; const float* ab2 = (const float*)d_in[13];
    const float* aw3 = (const float*)d_in[14]; const float* ab3 = (const float*)d_in[15];
    const float* sw1 = (const float*)d_in[16]; const float* sb1 = (const float*)d_in[17];
    const float* sw2 = (const float*)d_in[18]; const float* sb2 = (const float*)d_in[19];
    const float* sw3 = (const float*)d_in[20]; const float* sb3 = (const float*)d_in[21];
    const float* pw1 = (const float*)d_in[22]; const float* pb1 = (const float*)d_in[23];
    const float* pw2 = (const float*)d_in[24]; const float* pb2 = (const float*)d_in[25];
    const float* pw3 = (const float*)d_in[26]; const float* pb3 = (const float*)d_in[27];

    float* w  = (float*)d_ws;
    int*   iw = (int*)(w + OF_INT);
    int* keep = iw;
    int* kidx = iw + NS_;
    float* out = (float*)d_out;

    // 0) zero-padded weight/bias copies
    padcopy(aw1, SD_,   HID_, SD_,   w + OF_AW1, SD_,  NH_, stream);
    padcopy(aw2, HID_,  HID_, HID_,  w + OF_AW2, NH_,  NH_, stream);
    padcopy(sw1, SPAN_, HID_, SPAN_, w + OF_SW1, LDK_, NH_, stream);
    padcopy(sw2, HID_,  HID_, HID_,  w + OF_SW2, NH_,  NH_, stream);
    padcopy(pw1,              PAIRW_, HID_, SPAN_, w + OF_WA, LDK_, NH_, stream);
    padcopy(pw1 + SPAN_,      PAIRW_, HID_, SPAN_, w + OF_WB, LDK_, NH_, stream);
    padcopy(pw1 + 2 * SPAN_,  PAIRW_, HID_, SPAN_, w + OF_WC, LDK_, NH_, stream);
    padcopy(pw2, HID_,  HID_, HID_,  w + OF_PW2, NH_,  NH_, stream);
    padcopy(ab1, HID_, 1, HID_, w + OF_AB1, NH_, 1, stream);
    padcopy(ab2, HID_, 1, HID_, w + OF_AB2, NH_, 1, stream);
    padcopy(sb1, HID_, 1, HID_, w + OF_SB1, NH_, 1, stream);
    padcopy(sb2, HID_, 1, HID_, w + OF_SB2, NH_, 1, stream);
    padcopy(pb1, HID_, 1, HID_, w + OF_PB1, NH_, 1, stream);
    padcopy(pb2, HID_, 1, HID_, w + OF_PB2, NH_, 1, stream);
    padcopy(pw3, HID_, 1, HID_, w + OF_PW3, NH_, 1, stream);

    // 1) embedding gather
    gather_embed_kernel<<<T_LEN, 128, 0, stream>>>(tok, embed, w + OF_EMB);

    // 2) LSTM input pre-activations (768x300 @ 300x800; dims already WMMA-clean)
    dim3 gPre(T_LEN / 16, (G4_ + 127) / 128);
    gemm_wmma_kernel<<<gPre, 256, 0, stream>>>(w + OF_EMB, D_EMB, w_ih_f, D_EMB,
        w + OF_PREF, G4_, G4_, D_EMB, b_ih_f, b_hh_f, 0);
    gemm_wmma_kernel<<<gPre, 256, 0, stream>>>(w + OF_EMB, D_EMB, w_ih_r, D_EMB,
        w + OF_PRER, G4_, G4_, D_EMB, b_ih_r, b_hh_r, 0);

    // 3) sequential bi-LSTM
    lstm_kernel<<<2, 256, 0, stream>>>(w + OF_PREF, w + OF_PRER, w_hh_f, w_hh_r, w + OF_ST);

    // 4) attention MLP: a = MLP3(states)
    dim3 gA1(T_LEN / 16, (NH_ + 127) / 128);
    gemm_wmma_kernel<<<gA1, 256, 0, stream>>>(w + OF_ST, SD_, w + OF_AW1, SD_,
        w + OF_T1, NH_, NH_, SD_, w + OF_AB1, nullptr, 1);
    gemm_wmma_kernel<<<gA1, 256, 0, stream>>>(w + OF_T1, NH_, w + OF_AW2, NH_,
        w + OF_T2, NH_, NH_, NH_, w + OF_AB2, nullptr, 1);
    rowdot_kernel<<<(T_LEN + 255) / 256, 256, 0, stream>>>(w + OF_T2, NH_, aw3, ab3,
        w + OF_A, T_LEN, HID_);

    // 5) span features (grid includes pad rows)
    span_kernel<<<NSP_, 128, 0, stream>>>(w + OF_A, w + OF_EMB, w + OF_ST, w + OF_G);

    // 6) span MLP: sm = MLP3(g)
    dim3 gS1(NSP_ / 16, (NH_ + 127) / 128);
    gemm_wmma_kernel<<<gS1, 256, 0, stream>>>(w + OF_G, LDK_, w + OF_SW1, LDK_,
        w + OF_T1, NH_, NH_, LDK_, w + OF_SB1, nullptr, 1);
    gemm_wmma_kernel<<<gS1, 256, 0, stream>>>(w + OF_T1, NH_, w + OF_SW2, NH_,
        w + OF_T2, NH_, NH_, NH_, w + OF_SB2, nullptr, 1);
    rowdot_kernel<<<(NS_ + 255) / 256, 256, 0, stream>>>(w + OF_T2, NH_, sw3, sb3,
        w + OF_SM, NS_, HID_);

    // 7) top-k (rank count) + order by (start*T+end) + gather
    rank_kernel<<<(NS_ + 255) / 256, 256, 0, stream>>>(w + OF_SM, keep);
    scatter_kernel<<<(NS_ + 255) / 256, 256, 0, stream>>>(keep, kidx);
    gatherG_kernel<<<KKP_, 128, 0, stream>>>(w + OF_G, w + OF_SM, kidx,
        w + OF_GK, w + OF_SMK, out);

    // 8) A = G@wa^T, B = G@wb^T
    dim3 gAB(KKP_ / 16, (NH_ + 127) / 128);
    gemm_wmma_kernel<<<gAB, 256, 0, stream>>>(w + OF_GK, LDK_, w + OF_WA, LDK_,
        w + OF_AM, NH_, NH_, LDK_, nullptr, nullptr, 0);
    gemm_wmma_kernel<<<gAB, 256, 0, stream>>>(w + OF_GK, LDK_, w + OF_WB, LDK_,
        w + OF_BM, NH_, NH_, LDK_, nullptr, nullptr, 0);

    // 9) fused pairwise scorer -> lg
    dim3 gPair((KK_ + 15) / 16, (KK_ + 3) / 4);
    pair_kernel<<<gPair, 128, 0, stream>>>(w + OF_GK, w + OF_AM, w + OF_BM,
        w + OF_SMK, w + OF_WC, w + OF_PB1, w + OF_PW2, w + OF_PB2,
        w + OF_PW3, pb3, w + OF_LG);

    // 10) masked softmax (with appended zero column) -> pred
    softmax_kernel<<<KK_, 128, 0, stream>>>(w + OF_LG, out);
}